// Rule_67662914781186
// MI455X (gfx1250) — compile-verified
//
#include <hip/hip_runtime.h>
#include <math.h>

typedef __attribute__((ext_vector_type(2))) float v2f;
typedef __attribute__((ext_vector_type(8))) float v8f;

#define IMG 16
#define HH 1024
#define WW 1024
#define NPIX (IMG * HH * WW)            // 16,777,216 lattice sites
#define NBLK (IMG * HH)                 // 16,384 blocks (one row each)
#define STATE_ELEMS (IMG * 2 * HH * WW) // 33,554,432 floats of state output

// ---------------------------------------------------------------------------
// Kernel 1: fused stencil + Metropolis accept + flip + b-copy + block reduce.
// One block per (image,row); 256 threads x 4 px (float4). Memory-bound:
// coalesced 128-bit loads/stores; vertical neighbors hit L2 (192MB >> 64MB).
// ---------------------------------------------------------------------------
__global__ __launch_bounds__(256) void ising_row_kernel(
    const float* __restrict__ x, const float* __restrict__ rnd,
    const float* __restrict__ drop, float* __restrict__ out,
    float* __restrict__ ws)
{
    const int bid = blockIdx.x;
    const int img = bid >> 10;
    const int y   = bid & (HH - 1);
    const int tid = threadIdx.x;
    const int w0  = tid << 2;

    const float* srow = x + ((size_t)(img * 2) * HH + y) * WW;
    const float* brow = x + ((size_t)(img * 2 + 1) * HH + y) * WW;
    const float* urow = x + ((size_t)(img * 2) * HH + ((y + HH - 1) & (HH - 1))) * WW;
    const float* drow = x + ((size_t)(img * 2) * HH + ((y + 1) & (HH - 1))) * WW;
    const float* rrow = rnd + ((size_t)img * HH + y) * WW;
    const float* prow = drop + (size_t)y * WW;
    float* os = out + ((size_t)(img * 2) * HH + y) * WW;
    float* ob = out + ((size_t)(img * 2 + 1) * HH + y) * WW;

    const float4 c  = *(const float4*)(srow + w0);
    const float4 u  = *(const float4*)(urow + w0);
    const float4 dn = *(const float4*)(drow + w0);
    const float4 bb = *(const float4*)(brow + w0);
    const float4 rr = *(const float4*)(rrow + w0);
    const float4 dm = *(const float4*)(prow + w0);
    const float sL = srow[(w0 + WW - 1) & (WW - 1)];   // periodic wrap left
    const float sR = srow[(w0 + 4) & (WW - 1)];        // periodic wrap right

    const float sv[4] = {c.x, c.y, c.z, c.w};
    const float uv[4] = {u.x, u.y, u.z, u.w};
    const float dv[4] = {dn.x, dn.y, dn.z, dn.w};
    const float lv[4] = {sL, c.x, c.y, c.z};
    const float rv[4] = {c.y, c.z, c.w, sR};
    const float bv[4] = {bb.x, bb.y, bb.z, bb.w};
    const float rndv[4] = {rr.x, rr.y, rr.z, rr.w};
    const float dmv[4] = {dm.x, dm.y, dm.z, dm.w};

    float sum_de = 0.0f;  // sum of s * nbr
    float sum_s  = 0.0f;  // sum of s
    float res[4];
#pragma unroll
    for (int i = 0; i < 4; ++i) {
        const float nbr  = lv[i] + rv[i] + uv[i] + dv[i];
        const float prod = sv[i] * nbr;
        sum_de += prod;
        sum_s  += sv[i];
        const float de = 2.0f * prod;
        const float p  = (de <= 0.0f) ? 1.0f : expf(-de * bv[i]);
        const bool accept = (rndv[i] < p) && (dmv[i] > 0.5f);
        res[i] = accept ? -sv[i] : sv[i];
    }

    float4 outv;
    outv.x = res[0]; outv.y = res[1]; outv.z = res[2]; outv.w = res[3];
    *(float4*)(os + w0) = outv;   // flipped spin plane
    *(float4*)(ob + w0) = bb;     // b plane passes through

    // Deterministic block reduction (fp32 tree in LDS).
    __shared__ float s_de[256];
    __shared__ float s_s[256];
    s_de[tid] = sum_de;
    s_s[tid]  = sum_s;
    __syncthreads();
#pragma unroll
    for (int off = 128; off > 0; off >>= 1) {
        if (tid < off) {
            s_de[tid] += s_de[tid + off];
            s_s[tid]  += s_s[tid + off];
        }
        __syncthreads();
    }
    if (tid == 0) {
        ws[2 * bid]     = s_de[0];
        ws[2 * bid + 1] = s_s[0];
    }
}

// ---------------------------------------------------------------------------
// Kernel 2: one wave32 (EXEC all ones, as WMMA requires). Reduce 16,384 fp32
// partial pairs with V_WMMA_F32_16X16X4_F32 using an all-ones B matrix:
// D = A x ones + C sums 64 fp32 values per instruction at full precision.
// Each row-sum is replicated across 16 columns -> divide the grand total by 16.
// ---------------------------------------------------------------------------
__global__ __launch_bounds__(32) void reduce_wmma_kernel(
    const float* __restrict__ ws, float* __restrict__ out)
{
    const int lane = threadIdx.x;

    v8f cde = {};
    v8f cs  = {};
    v2f ones;
    ones.x = 1.0f; ones.y = 1.0f;

    for (int it = 0; it < NBLK / 64; ++it) {
        const int idx = it * 64 + lane * 2;   // two A elements per lane
        v2f ade, as;
        ade.x = ws[2 * idx];
        ade.y = ws[2 * (idx + 1)];
        as.x  = ws[2 * idx + 1];
        as.y  = ws[2 * (idx + 1) + 1];
        // 8 args: (neg_a, A, neg_b, B, c_mod, C, reuse_a, reuse_b)
        cde = __builtin_amdgcn_wmma_f32_16x16x4_f32(
            false, ade, false, ones, (short)0, cde, false, false);
        cs = __builtin_amdgcn_wmma_f32_16x16x4_f32(
            false, as, false, ones, (short)0, cs, false, false);
    }

    float tde = 0.0f, ts = 0.0f;
#pragma unroll
    for (int i = 0; i < 8; ++i) { tde += cde[i]; ts += cs[i]; }

    __shared__ float rde[32];
    __shared__ float rs[32];
    rde[lane] = tde;
    rs[lane]  = ts;
    __syncthreads();

    if (lane == 0) {
        float gde = 0.0f, gs = 0.0f;
        for (int i = 0; i < 32; ++i) { gde += rde[i]; gs += rs[i]; }
        gde *= (1.0f / 16.0f);   // undo 16x column replication
        gs  *= (1.0f / 16.0f);
        const float invN = 1.0f / (float)NPIX;
        const float e = -gde * invN * 0.25f;  // mean(-0.5*2*s*nbr)/4
        const float m = gs * invN;
        out[STATE_ELEMS + 0] = e;
        out[STATE_ELEMS + 1] = e * e;
        out[STATE_ELEMS + 2] = m;
        out[STATE_ELEMS + 3] = m * m;
    }
}

extern "C" void kernel_launch(void* const* d_in, const int* in_sizes, int n_in,
                              void* d_out, int out_size, void* d_ws, size_t ws_size,
                              hipStream_t stream) {
    const float* x    = (const float*)d_in[0];   // (16, 2, 1024, 1024) f32
    const float* rnd  = (const float*)d_in[1];   // (16, 1, 1024, 1024) f32
    const float* drop = (const float*)d_in[2];   // (1024, 1024) f32
    float* out = (float*)d_out;                  // state ++ obvs
    float* ws  = (float*)d_ws;                   // 2 * NBLK floats of partials

    ising_row_kernel<<<NBLK, 256, 0, stream>>>(x, rnd, drop, out, ws);
    reduce_wmma_kernel<<<1, 32, 0, stream>>>(ws, out);
}